// GraphConvolution_28991029248866
// MI455X (gfx1250) — compile-verified
//
#include <hip/hip_runtime.h>

#define N_NODES 100000
#define N_EDGES 1600000
#define IN_F    256
#define OUT_F   128
#define LDS_STRIDE 260     // 256 + 4 pad: 1040B row pitch, 16B aligned, bank-conflict-free
#define NCHUNK  98         // ceil(N_NODES / 1024)

typedef __attribute__((ext_vector_type(2))) float v2f;
typedef __attribute__((ext_vector_type(8))) float v8f;

static __device__ __forceinline__ int atomic_add_i32(int* p, int v) {
    return __hip_atomic_fetch_add(p, v, __ATOMIC_RELAXED, __HIP_MEMORY_SCOPE_AGENT);
}

// ---------------------------------------------------------------------------
// 1) counts[i] = 0
// ---------------------------------------------------------------------------
__global__ __launch_bounds__(256) void k_zero(int* __restrict__ counts) {
    int i = blockIdx.x * blockDim.x + threadIdx.x;
    if (i < N_NODES) counts[i] = 0;
}

// ---------------------------------------------------------------------------
// 2) histogram of edge_row (1.6M int atomics)
// ---------------------------------------------------------------------------
__global__ __launch_bounds__(256) void k_hist(const int* __restrict__ erow,
                                              int* __restrict__ counts) {
    int e = blockIdx.x * blockDim.x + threadIdx.x;   // grid sized exactly
    atomic_add_i32(&counts[erow[e]], 1);
}

// ---------------------------------------------------------------------------
// 3) per-1024-chunk sums: blockSums[b] = sum(counts[b*1024 .. b*1024+1023])
// ---------------------------------------------------------------------------
__global__ __launch_bounds__(256) void k_blocksum(const int* __restrict__ counts,
                                                  int* __restrict__ blockSums) {
    __shared__ int sb[256];
    int t = threadIdx.x;
    int base = blockIdx.x * 1024 + t * 4;
    int s = 0;
    #pragma unroll
    for (int j = 0; j < 4; ++j) {
        int i = base + j;
        s += (i < N_NODES) ? counts[i] : 0;
    }
    sb[t] = s; __syncthreads();
    for (int off = 128; off > 0; off >>= 1) {
        if (t < off) sb[t] += sb[t + off];
        __syncthreads();
    }
    if (t == 0) blockSums[blockIdx.x] = sb[0];
}

// ---------------------------------------------------------------------------
// 4) single-block exclusive scan of NCHUNK block sums (in place);
//    also start[N_NODES] = N_EDGES
// ---------------------------------------------------------------------------
__global__ __launch_bounds__(128) void k_scantop(int* __restrict__ blockSums,
                                                 int* __restrict__ start) {
    __shared__ int sb[128];
    int t = threadIdx.x;
    int v = (t < NCHUNK) ? blockSums[t] : 0;
    sb[t] = v; __syncthreads();
    for (int off = 1; off < 128; off <<= 1) {
        int x = (t >= off) ? sb[t - off] : 0;
        __syncthreads();
        sb[t] += x;
        __syncthreads();
    }
    if (t < NCHUNK) blockSums[t] = sb[t] - v;        // exclusive prefix
    if (t == 0) start[N_NODES] = N_EDGES;
}

// ---------------------------------------------------------------------------
// 5) per-chunk exclusive scan + global offset -> start[], cursor[]
// ---------------------------------------------------------------------------
__global__ __launch_bounds__(256) void k_scanchunk(const int* __restrict__ counts,
                                                   const int* __restrict__ blockSums,
                                                   int* __restrict__ start,
                                                   int* __restrict__ cursor) {
    __shared__ int sb[256];
    int t = threadIdx.x;
    int base = blockIdx.x * 1024 + t * 4;
    int v[4]; int s = 0;
    #pragma unroll
    for (int j = 0; j < 4; ++j) {
        int i = base + j;
        v[j] = (i < N_NODES) ? counts[i] : 0;
        s += v[j];
    }
    sb[t] = s; __syncthreads();
    for (int off = 1; off < 256; off <<= 1) {
        int x = (t >= off) ? sb[t - off] : 0;
        __syncthreads();
        sb[t] += x;
        __syncthreads();
    }
    int run = blockSums[blockIdx.x] + sb[t] - s;     // exclusive base for this thread
    #pragma unroll
    for (int j = 0; j < 4; ++j) {
        int i = base + j;
        if (i < N_NODES) { start[i] = run; cursor[i] = run; }
        run += v[j];
    }
}

// ---------------------------------------------------------------------------
// 6) counting-sort scatter: spair[p] = (col, val) packed int2
// ---------------------------------------------------------------------------
__global__ __launch_bounds__(256) void k_sort(const int*   __restrict__ erow,
                                              const int*   __restrict__ ecol,
                                              const float* __restrict__ eval,
                                              int* __restrict__ cursor,
                                              int2* __restrict__ spair) {
    int e = blockIdx.x * blockDim.x + threadIdx.x;   // grid sized exactly
    int r = erow[e];
    int p = atomic_add_i32(&cursor[r], 1);
    spair[p] = make_int2(ecol[e], __float_as_int(eval[e]));
}

// ---------------------------------------------------------------------------
// 7) support = X @ W  via V_WMMA_F32_16X16X4_F32 (unchanged; lowering verified)
// ---------------------------------------------------------------------------
__global__ __launch_bounds__(256) void gcn_gemm(const float* __restrict__ X,
                                                const float* __restrict__ W,
                                                float* __restrict__ support) {
    __shared__ float xs[16 * LDS_STRIDE];

    const int row0 = blockIdx.x * 16;
    const int tid  = threadIdx.x;

    for (int i = tid; i < 16 * (IN_F / 4); i += 256) {
        int r  = i >> 6;
        int c4 = i & 63;
        float4 v = *(const float4*)(X + (size_t)(row0 + r) * IN_F + c4 * 4);
        *(float4*)(&xs[r * LDS_STRIDE + c4 * 4]) = v;
    }
    __syncthreads();

    const int wave = tid >> 5;
    const int lane = tid & 31;
    const int m    = lane & 15;
    const int kk   = (lane >> 4) * 2;
    const int n    = (lane & 15) + wave * 16;

    v8f acc = {};
    #pragma unroll
    for (int k = 0; k < IN_F; k += 4) {
        v2f a = *(const v2f*)(&xs[m * LDS_STRIDE + k + kk]);
        v2f b;
        b.x = W[(k + kk)     * OUT_F + n];
        b.y = W[(k + kk + 1) * OUT_F + n];
        acc = __builtin_amdgcn_wmma_f32_16x16x4_f32(
            false, a, false, b, (short)0, acc, false, false);
    }

    const int mBase = (lane >> 4) * 8;
    #pragma unroll
    for (int i = 0; i < 8; ++i) {
        support[(size_t)(row0 + mBase + i) * OUT_F + n] = acc[i];
    }
}

// ---------------------------------------------------------------------------
// 8) atomic-free segment sum: one wave per row, bias fused
// ---------------------------------------------------------------------------
__global__ __launch_bounds__(256) void k_segsum(const int*   __restrict__ start,
                                                const int2*  __restrict__ spair,
                                                const float* __restrict__ support,
                                                const float* __restrict__ bias,
                                                float* __restrict__ out) {
    int wid  = (blockIdx.x * 256 + threadIdx.x) >> 5;   // row id; grid sized exactly
    int lane = threadIdx.x & 31;

    int e0 = start[wid];
    int e1 = start[wid + 1];

    float4 acc = *(const float4*)(bias + lane * 4);
    for (int e = e0; e < e1; ++e) {
        int2  p = spair[e];                              // one b64 per edge (uniform, L2)
        float v = __int_as_float(p.y);
        float4 s = *(const float4*)(support + (size_t)p.x * OUT_F + lane * 4);
        acc.x += v * s.x; acc.y += v * s.y; acc.z += v * s.z; acc.w += v * s.w;
    }
    *(float4*)(out + (size_t)wid * OUT_F + lane * 4) = acc;
}

// ---------------------------------------------------------------------------
// Launch. d_ws layout (bytes, all 16B aligned):
//   [0, 51.2M)          support  (fp32, stays L2-resident)
//   [51.2M, +400K)      counts
//   [51.6M, +400K+4)    start (N_NODES+1)
//   [52.000256M, +400K) cursor
//   [52.400256M, +512)  blockSums (NCHUNK)
//   [52.400768M, +12.8M) spair (int2 per edge)
// total ~65.2 MB
// ---------------------------------------------------------------------------
extern "C" void kernel_launch(void* const* d_in, const int* in_sizes, int n_in,
                              void* d_out, int out_size, void* d_ws, size_t ws_size,
                              hipStream_t stream) {
    const float* x    = (const float*)d_in[0];
    const int*   erow = (const int*)  d_in[1];
    const int*   ecol = (const int*)  d_in[2];
    const float* eval = (const float*)d_in[3];
    const float* w    = (const float*)d_in[4];
    const float* bias = (const float*)d_in[5];
    float* out        = (float*)d_out;

    char* ws = (char*)d_ws;
    float* support   = (float*)(ws);
    int*   counts    = (int*)  (ws + 51200000);
    int*   start     = (int*)  (ws + 51600000);
    int*   cursor    = (int*)  (ws + 52000256);
    int*   blockSums = (int*)  (ws + 52400256);
    int2*  spair     = (int2*) (ws + 52400768);

    k_zero     <<<(N_NODES + 255) / 256, 256, 0, stream>>>(counts);
    k_hist     <<<N_EDGES / 256,         256, 0, stream>>>(erow, counts);
    k_blocksum <<<NCHUNK,                256, 0, stream>>>(counts, blockSums);
    k_scantop  <<<1,                     128, 0, stream>>>(blockSums, start);
    k_scanchunk<<<NCHUNK,                256, 0, stream>>>(counts, blockSums, start, cursor);
    k_sort     <<<N_EDGES / 256,         256, 0, stream>>>(erow, ecol, eval, cursor, spair);
    gcn_gemm   <<<N_NODES / 16,          256, 0, stream>>>(x, w, support);
    k_segsum   <<<(N_NODES * 32) / 256,  256, 0, stream>>>(start, spair, support, bias, out);
}